// BaselineGNNLSTM_85418309582938
// MI455X (gfx1250) — compile-verified
//
#include <hip/hip_runtime.h>
#include <hip/hip_bf16.h>

// ---------------- problem constants (from reference) ----------------
constexpr int kB  = 32;
constexpr int kT  = 12;
constexpr int kNP = 2000;
constexpr int kN  = kB * kNP;      // 64000 nodes
constexpr int kE  = 1024000;       // edges
constexpr int kGH = 32;            // conv hidden
constexpr int kNO = 8;             // lstm input
constexpr int kH  = 64;            // lstm hidden
constexpr int kWF = 4;
constexpr int kTF = 8;
constexpr int kAC = 64;            // acat row stride = [agg2(32) | h1(32)]

typedef __attribute__((ext_vector_type(2))) float v2f;
typedef __attribute__((ext_vector_type(8))) float v8f;

__device__ __forceinline__ float sigmoidf_(float x) {
    return 1.0f / (1.0f + __expf(-x));
}

// ---- stage 0: build xN (last time slice), zero agg1 and the acat buffer ----
__global__ __launch_bounds__(256) void init_kernel(const float* __restrict__ x,
                                                   float* __restrict__ xN,
                                                   float* __restrict__ agg1,
                                                   float* __restrict__ acat) {
    int i = blockIdx.x * 256 + threadIdx.x;          // 0 .. 64*kN
    if (i < kAC * kN) acat[i] = 0.0f;
    if (i < kN) {
        int b = i / kNP, p = i % kNP;
        xN[i]   = x[(b * kT + (kT - 1)) * kNP + p];  // x[b, T-1, p]
        agg1[i] = 0.0f;
    }
}

// ---- stage 0b: pack B = [W2rel; W2root] (64x8) into column-major padded 16x64 ----
__global__ __launch_bounds__(256) void pack_w2_kernel(const float* __restrict__ W2rel,
                                                      const float* __restrict__ W2root,
                                                      float* __restrict__ w2pad) {
    int i = blockIdx.x * 256 + threadIdx.x;          // 0 .. 1024
    if (i >= 16 * kAC) return;
    int j = i >> 6;                                  // output column 0..15
    int k = i & 63;                                  // K index 0..63
    float v = 0.0f;
    if (j < kNO) v = (k < kGH) ? W2rel[k * kNO + j] : W2root[(k - kGH) * kNO + j];
    w2pad[j * kAC + k] = v;                          // column-major: contiguous in K
}

// ---- stage 1: scalar edge scatter: agg1[dst] += xN[src] ----
__global__ __launch_bounds__(256) void scatter1_kernel(const int* __restrict__ edge,
                                                       const float* __restrict__ xN,
                                                       float* __restrict__ agg1) {
    int e = blockIdx.x * 256 + threadIdx.x;
    if (e >= kE) return;
    int s = edge[e];          // src row
    int d = edge[kE + e];     // dst row
    atomicAdd(&agg1[d], xN[s]);
}

// ---- stage 2: h1 half of acat: acat[n][32+f] = relu(agg1[n]*W1rel + b1 + xN[n]*W1root) ----
__global__ __launch_bounds__(256) void h1_kernel(const float* __restrict__ agg1,
                                                 const float* __restrict__ xN,
                                                 const float* __restrict__ W1rel,
                                                 const float* __restrict__ b1,
                                                 const float* __restrict__ W1root,
                                                 float* __restrict__ acat) {
    int i = blockIdx.x * 256 + threadIdx.x;
    if (i >= kN * kGH) return;
    int n = i >> 5, f = i & 31;
    float v = agg1[n] * W1rel[f] + b1[f] + xN[n] * W1root[f];
    acat[n * kAC + kGH + f] = v > 0.0f ? v : 0.0f;
}

// ---- stage 3: 32-wide edge scatter: acat[dst][0:32] += acat[src][32:64] ----
__global__ __launch_bounds__(256) void scatter2_kernel(const int* __restrict__ edge,
                                                       float* __restrict__ acat) {
    long i = (long)blockIdx.x * 256 + threadIdx.x;   // 0 .. E*32
    if (i >= (long)kE * kGH) return;
    int e = (int)(i >> 5);
    int f = (int)(i & 31);
    int s = edge[e];
    int d = edge[kE + e];
    atomicAdd(&acat[d * kAC + f], acat[s * kAC + kGH + f]);
}

// ---- stage 4: lstm_in = acat (16x64 tile) @ w2pad (64x16, 8 valid cols) + b2
//      fp32 WMMA (v_wmma_f32_16x16x4_f32), branch-free b64 fragment loads.
__global__ __launch_bounds__(32) void lstmin_wmma_kernel(const float* __restrict__ acat,
                                                         const float* __restrict__ w2pad,
                                                         const float* __restrict__ b2,
                                                         float* __restrict__ lstm_in) {
    const int lane  = threadIdx.x;     // 0..31
    const int m     = lane & 15;       // A row / B (=C) column
    const int khalf = lane >> 4;       // K pair within a 4-chunk
    const int node0 = blockIdx.x * 16;

    const float* arow = acat + (long)(node0 + m) * kAC + khalf * 2;  // A: row m
    const float* brow = w2pad + m * kAC + khalf * 2;                 // B: column m

    // Preload all B fragments (16 x global_load_b64, immediate offsets).
    v2f bfrag[16];
#pragma unroll
    for (int kk = 0; kk < 16; ++kk)
        bfrag[kk] = *(const v2f*)(brow + kk * 4);

    v8f c = {};
#pragma unroll
    for (int kk = 0; kk < 16; ++kk) {
        v2f a = *(const v2f*)(arow + kk * 4);
        c = __builtin_amdgcn_wmma_f32_16x16x4_f32(
            /*neg_a=*/false, a, /*neg_b=*/false, bfrag[kk],
            /*c_mod=*/(short)0, c, /*reuse_a=*/false, /*reuse_b=*/false);
    }

    // C layout: VGPR r, lanes 0-15 -> (M=r, Ncol=lane); lanes 16-31 -> (M=r+8, Ncol=lane-16)
    if (m < kNO) {
        float bias = b2[m];
#pragma unroll
        for (int r = 0; r < 8; ++r) {
            int mm = r + khalf * 8;
            lstm_in[(node0 + mm) * kNO + m] = c[r] + bias;
        }
    }
}

// ---- stage 5: the serial LSTM chain over 64000 nodes (batch row T-1 only).
//      256 lanes; lane j owns gate column j with its 64 Whh weights in VGPRs.
//      Only h (64 floats) is broadcast via LDS; 2 barriers/step.
__global__ __launch_bounds__(256) void lstm_kernel(const float* __restrict__ lstm_in,
                                                   const float* __restrict__ Wih,
                                                   const float* __restrict__ Whh,
                                                   const float* __restrict__ bih,
                                                   const float* __restrict__ bhh,
                                                   float* __restrict__ outs) {
    __shared__ alignas(16) float sh[kH];     // hidden state broadcast
    __shared__ float sg[4 * kH];             // gate pre-activations
    __shared__ float sx[2][kNO];             // double-buffered input row

    const int tid = threadIdx.x;             // 0..255, gate column
    float wih_r[kNO];
    float whh_r[kH];
#pragma unroll
    for (int k = 0; k < kNO; ++k) wih_r[k] = Wih[k * 256 + tid];
#pragma unroll
    for (int k = 0; k < kH; ++k)  whh_r[k] = Whh[k * 256 + tid];
    const float bsum = bih[tid] + bhh[tid];

    if (tid < kH) sh[tid] = 0.0f;
    if (tid < kNO) sx[0][tid] = lstm_in[tid];
    float creg = 0.0f;                       // lanes 0..63 own c
    __syncthreads();

    const float4* sh4 = reinterpret_cast<const float4*>(sh);
    for (int n = 0; n < kN; ++n) {
        const int cur = n & 1;
        float g = bsum;
#pragma unroll
        for (int k = 0; k < kNO; ++k) g += sx[cur][k] * wih_r[k];
#pragma unroll
        for (int k4 = 0; k4 < kH / 4; ++k4) {
            float4 hv = sh4[k4];             // ds_load_b128 broadcast
            g += hv.x * whh_r[4 * k4 + 0] + hv.y * whh_r[4 * k4 + 1] +
                 hv.z * whh_r[4 * k4 + 2] + hv.w * whh_r[4 * k4 + 3];
        }
        sg[tid] = g;
        __syncthreads();                     // gates ready; h/x reads of step n done

        if (tid < kNO && (n + 1) < kN) sx[cur ^ 1][tid] = lstm_in[(n + 1) * kNO + tid];
        if (tid < kH) {
            float iv = sigmoidf_(sg[tid]);
            float fv = sigmoidf_(sg[kH + tid]);
            float gv = tanhf(sg[2 * kH + tid]);
            float ov = sigmoidf_(sg[3 * kH + tid]);
            creg = fv * creg + iv * gv;
            float hn = ov * tanhf(creg);
            sh[tid] = hn;
            outs[(long)n * kH + tid] = hn;
        }
        __syncthreads();                     // h/x ready for step n+1
    }
}

// ---- stage 6: pred[n] = h_n . Wlin[0:64] + weather[n%B,-1] . Wlin[64:68]
//                       + time[n%B,-1] . Wlin[68:76] + blin
__global__ __launch_bounds__(256) void epilogue_kernel(const float* __restrict__ outs,
                                                       const float* __restrict__ weather,
                                                       const float* __restrict__ time_enc,
                                                       const float* __restrict__ Wlin,
                                                       const float* __restrict__ blin,
                                                       float* __restrict__ out) {
    int n = blockIdx.x * 256 + threadIdx.x;
    if (n >= kN) return;
    float acc = blin[0];
    const float* hrow = outs + (long)n * kH;
#pragma unroll
    for (int j = 0; j < kH; ++j) acc += hrow[j] * Wlin[j];
    int bb = n % kB;                         // jnp.tile((B,WF),(NP,1)) -> row n % B
    const float* wrow = weather  + (bb * kT + (kT - 1)) * kWF;
    const float* trow = time_enc + (bb * kT + (kT - 1)) * kTF;
#pragma unroll
    for (int k = 0; k < kWF; ++k) acc += wrow[k] * Wlin[kH + k];
#pragma unroll
    for (int k = 0; k < kTF; ++k) acc += trow[k] * Wlin[kH + kWF + k];
    out[n] = acc;
}

// ---------------------------------------------------------------------------
extern "C" void kernel_launch(void* const* d_in, const int* in_sizes, int n_in,
                              void* d_out, int out_size, void* d_ws, size_t ws_size,
                              hipStream_t stream) {
    const float* x        = (const float*)d_in[0];
    const int*   edge     = (const int*)  d_in[1];
    const float* weather  = (const float*)d_in[2];
    const float* time_enc = (const float*)d_in[3];
    const float* W1rel    = (const float*)d_in[4];
    const float* b1       = (const float*)d_in[5];
    const float* W1root   = (const float*)d_in[6];
    const float* W2rel    = (const float*)d_in[7];
    const float* b2       = (const float*)d_in[8];
    const float* W2root   = (const float*)d_in[9];
    const float* Wih      = (const float*)d_in[10];
    const float* Whh      = (const float*)d_in[11];
    const float* bih      = (const float*)d_in[12];
    const float* bhh      = (const float*)d_in[13];
    const float* Wlin     = (const float*)d_in[14];
    const float* blin     = (const float*)d_in[15];
    float* out = (float*)d_out;

    // Workspace layout (floats). outs aliases acat (dead after the WMMA stage).
    float* ws      = (float*)d_ws;
    float* xN      = ws;                     // N
    float* agg1    = ws + kN;                // N
    float* acat    = ws + 2 * kN;            // 64N: [agg2(32) | h1(32)] per node
    float* outs    = ws + 2 * kN;            // 64N (aliases acat)
    float* lstm_in = ws + 66 * kN;           // 8N
    float* w2pad   = ws + 74 * kN;           // 1024  -> total ~18.9 MB

    init_kernel<<<(kAC * kN + 255) / 256, 256, 0, stream>>>(x, xN, agg1, acat);
    pack_w2_kernel<<<(16 * kAC + 255) / 256, 256, 0, stream>>>(W2rel, W2root, w2pad);
    scatter1_kernel<<<(kE + 255) / 256, 256, 0, stream>>>(edge, xN, agg1);
    h1_kernel<<<(kN * kGH + 255) / 256, 256, 0, stream>>>(agg1, xN, W1rel, b1, W1root, acat);
    scatter2_kernel<<<(int)(((long)kE * kGH + 255) / 256), 256, 0, stream>>>(edge, acat);
    lstmin_wmma_kernel<<<kN / 16, 32, 0, stream>>>(acat, w2pad, b2, lstm_in);
    lstm_kernel<<<1, 256, 0, stream>>>(lstm_in, Wih, Whh, bih, bhh, outs);
    epilogue_kernel<<<(kN + 255) / 256, 256, 0, stream>>>(outs, weather, time_enc, Wlin, blin, out);
}